// MoEGate_33200097198619
// MI455X (gfx1250) — compile-verified
//
#include <hip/hip_runtime.h>

typedef __attribute__((ext_vector_type(2))) float v2f;
typedef __attribute__((ext_vector_type(8))) float v8f;

#define NTOK  16384
#define HID   4096
#define NEXP  64
#define TOPK  6
#define WAVES_PER_BLOCK 2
#define TOKS_PER_WAVE   32   // two 16-row A tiles per wave

// swap helper for the sorted top-6 insertion
#define CSWAP(va, ia, vb, ib)                         \
  do {                                                \
    if ((vb) > (va)) {                                \
      float _tv = (va); (va) = (vb); (vb) = _tv;      \
      int   _ti = (ia); (ia) = (ib); (ib) = _ti;      \
    }                                                 \
  } while (0)

__launch_bounds__(WAVES_PER_BLOCK * 32, 1)
__global__ void moe_gate_kernel(const float* __restrict__ x,
                                const float* __restrict__ w,
                                int*   __restrict__ out_idx,
                                float* __restrict__ out_w) {
  // padded LDS logits: [wave][token 0..31][expert 0..63], 65-float pitch
  __shared__ float lds_logits[WAVES_PER_BLOCK * TOKS_PER_WAVE * (NEXP + 1)];

  const int lane  = threadIdx.x & 31;
  const int wave  = threadIdx.x >> 5;
  const int lrow  = lane & 15;   // token row (A) / expert col (B) within tile
  const int lhalf = lane >> 4;   // 0: K={0,1}, 1: K={2,3}

  const int tok0 = (blockIdx.x * WAVES_PER_BLOCK + wave) * TOKS_PER_WAVE;

  // A operands: two 16-token tiles of X
  const float* __restrict__ xr0 = x + (size_t)(tok0 + lrow)      * HID + 2 * lhalf;
  const float* __restrict__ xr1 = x + (size_t)(tok0 + 16 + lrow) * HID + 2 * lhalf;
  // B operands: 4 tiles of 16 experts of W
  const float* __restrict__ w0 = w + (size_t)(lrow +  0) * HID + 2 * lhalf;
  const float* __restrict__ w1 = w + (size_t)(lrow + 16) * HID + 2 * lhalf;
  const float* __restrict__ w2 = w + (size_t)(lrow + 32) * HID + 2 * lhalf;
  const float* __restrict__ w3 = w + (size_t)(lrow + 48) * HID + 2 * lhalf;

  // accumulators: [m-tile][n-tile], 8 x v8f = 64 VGPRs
  v8f a00 = {}, a01 = {}, a02 = {}, a03 = {};
  v8f a10 = {}, a11 = {}, a12 = {}, a13 = {};

#pragma unroll 8
  for (int k = 0; k < HID; k += 4) {
    if ((k & 31) == 0) {
      // pull both A rows ~4KB ahead (emits global_prefetch_b8)
      __builtin_prefetch(xr0 + k + 1024, 0, 1);
      __builtin_prefetch(xr1 + k + 1024, 0, 1);
    }
    v2f va0 = *(const v2f*)(xr0 + k);
    v2f va1 = *(const v2f*)(xr1 + k);
    v2f b0  = *(const v2f*)(w0 + k);
    v2f b1  = *(const v2f*)(w1 + k);
    v2f b2  = *(const v2f*)(w2 + k);
    v2f b3  = *(const v2f*)(w3 + k);
    // 8 WMMAs per 6 loads: each B tile reused across both A tiles
    a00 = __builtin_amdgcn_wmma_f32_16x16x4_f32(false, va0, false, b0, (short)0, a00, false, false);
    a10 = __builtin_amdgcn_wmma_f32_16x16x4_f32(false, va1, false, b0, (short)0, a10, false, false);
    a01 = __builtin_amdgcn_wmma_f32_16x16x4_f32(false, va0, false, b1, (short)0, a01, false, false);
    a11 = __builtin_amdgcn_wmma_f32_16x16x4_f32(false, va1, false, b1, (short)0, a11, false, false);
    a02 = __builtin_amdgcn_wmma_f32_16x16x4_f32(false, va0, false, b2, (short)0, a02, false, false);
    a12 = __builtin_amdgcn_wmma_f32_16x16x4_f32(false, va1, false, b2, (short)0, a12, false, false);
    a03 = __builtin_amdgcn_wmma_f32_16x16x4_f32(false, va0, false, b3, (short)0, a03, false, false);
    a13 = __builtin_amdgcn_wmma_f32_16x16x4_f32(false, va1, false, b3, (short)0, a13, false, false);
  }

  // Spill logits tiles to LDS.
  // C/D layout: VGPR r, lanes 0-15 -> token M=r, lanes 16-31 -> token M=r+8;
  // N (expert within tile) = lane & 15.
  float* myl = lds_logits + wave * TOKS_PER_WAVE * (NEXP + 1);
#pragma unroll
  for (int r = 0; r < 8; ++r) {
    const int m0 = r + 8 * lhalf;        // tokens 0..15 tile
    const int m1 = 16 + m0;              // tokens 16..31 tile
    myl[m0 * (NEXP + 1) + lrow +  0] = a00[r];
    myl[m0 * (NEXP + 1) + lrow + 16] = a01[r];
    myl[m0 * (NEXP + 1) + lrow + 32] = a02[r];
    myl[m0 * (NEXP + 1) + lrow + 48] = a03[r];
    myl[m1 * (NEXP + 1) + lrow +  0] = a10[r];
    myl[m1 * (NEXP + 1) + lrow + 16] = a11[r];
    myl[m1 * (NEXP + 1) + lrow + 32] = a12[r];
    myl[m1 * (NEXP + 1) + lrow + 48] = a13[r];
  }
  __syncthreads();

  // Softmax + top-6: every lane owns one of the wave's 32 tokens.
  {
    const float* lg = myl + lane * (NEXP + 1);

    // pass 1: max logit
    float mx = lg[0];
#pragma unroll 8
    for (int e = 1; e < NEXP; ++e) mx = fmaxf(mx, lg[e]);

    // pass 2: softmax denominator + sorted top-6 insertion on logits
    float s = 0.0f;
    float bv0 = -3.0e38f, bv1 = -3.0e38f, bv2 = -3.0e38f,
          bv3 = -3.0e38f, bv4 = -3.0e38f, bv5 = -3.0e38f;
    int   bi0 = 0, bi1 = 0, bi2 = 0, bi3 = 0, bi4 = 0, bi5 = 0;
#pragma unroll 4
    for (int e = 0; e < NEXP; ++e) {
      const float v = lg[e];
      s += __expf(v - mx);
      if (v > bv5) {
        bv5 = v; bi5 = e;
        CSWAP(bv4, bi4, bv5, bi5);
        CSWAP(bv3, bi3, bv4, bi4);
        CSWAP(bv2, bi2, bv3, bi3);
        CSWAP(bv1, bi1, bv2, bi2);
        CSWAP(bv0, bi0, bv1, bi1);
      }
    }

    const float inv_s = 1.0f / s;
    float e0 = __expf(bv0 - mx) * inv_s;
    float e1 = __expf(bv1 - mx) * inv_s;
    float e2 = __expf(bv2 - mx) * inv_s;
    float e3 = __expf(bv3 - mx) * inv_s;
    float e4 = __expf(bv4 - mx) * inv_s;
    float e5 = __expf(bv5 - mx) * inv_s;
    const float den = e0 + e1 + e2 + e3 + e4 + e5 + 1e-20f;
    const float inv_den = 1.0f / den;  // ROUTED_SCALING_FACTOR == 1.0

    const int g = tok0 + lane;
    int*   oi = out_idx + (size_t)g * TOPK;
    float* ow = out_w   + (size_t)g * TOPK;
    oi[0] = bi0; oi[1] = bi1; oi[2] = bi2; oi[3] = bi3; oi[4] = bi4; oi[5] = bi5;
    ow[0] = e0 * inv_den; ow[1] = e1 * inv_den; ow[2] = e2 * inv_den;
    ow[3] = e3 * inv_den; ow[4] = e4 * inv_den; ow[5] = e5 * inv_den;
  }
}

extern "C" void kernel_launch(void* const* d_in, const int* in_sizes, int n_in,
                              void* d_out, int out_size, void* d_ws, size_t ws_size,
                              hipStream_t stream) {
  (void)in_sizes; (void)n_in; (void)out_size; (void)d_ws; (void)ws_size;
  const float* x = (const float*)d_in[0];   // [16384, 4096] f32
  const float* w = (const float*)d_in[1];   // [64, 4096] f32
  // d_out = topk_idx (int32, 16384*6) followed by topk_weight (f32, 16384*6)
  int*   out_idx = (int*)d_out;
  float* out_w   = (float*)d_out + (size_t)NTOK * TOPK;

  const int tokens_per_block = WAVES_PER_BLOCK * TOKS_PER_WAVE;  // 64
  dim3 grid(NTOK / tokens_per_block);                            // 256
  dim3 block(WAVES_PER_BLOCK * 32);                              // 64
  moe_gate_kernel<<<grid, block, 0, stream>>>(x, w, out_idx, out_w);
}